// Selective_Attention_28630251995247
// MI455X (gfx1250) — compile-verified
//
#include <hip/hip_runtime.h>
#include <hip/hip_bf16.h>

// Problem constants (fixed in the reference file).
constexpr int N_SENT  = 65536;
constexpr int N_PAIRS = 8192;
constexpr int D       = 690;
constexpr int CLS     = 53;
constexpr int NPAD    = 64;    // classes padded to 64 (4 WMMA column tiles)
constexpr int KPAD    = 704;   // K padded to multiple of 4
constexpr int MAXC    = 1024;  // bag-size cap for LDS staging (actual max ~25)

typedef float v2f __attribute__((ext_vector_type(2)));
typedef float v8f __attribute__((ext_vector_type(8)));

// ---------------------------------------------------------------------------
// Kernel 1: zero the histogram, build zero-padded transposed-access embedding
// embPad[n*KPAD + k] = (n < CLS && k < D) ? emb[n*D + k] : 0
// ---------------------------------------------------------------------------
__global__ void prep_kernel(const float* __restrict__ emb,
                            float* __restrict__ embPad,
                            int* __restrict__ counts) {
  int idx = blockIdx.x * blockDim.x + threadIdx.x;
  if (idx < NPAD * KPAD) {
    int n = idx / KPAD, k = idx - n * KPAD;
    embPad[idx] = (n < CLS && k < D) ? emb[n * D + k] : 0.0f;
  }
  if (idx < N_PAIRS) counts[idx] = 0;
}

// ---------------------------------------------------------------------------
// Kernel 2: per-bag histogram
// ---------------------------------------------------------------------------
__global__ void count_kernel(const long long* __restrict__ epid,
                             int* __restrict__ counts) {
  int s = blockIdx.x * blockDim.x + threadIdx.x;
  if (s < N_SENT) atomicAdd(&counts[(int)epid[s]], 1);
}

// ---------------------------------------------------------------------------
// Kernel 3: exclusive scan of 8192 counts (single 1024-thread block, 8/thread)
// writes offsets and initializes cursors = offsets
// ---------------------------------------------------------------------------
__global__ void scan_kernel(const int* __restrict__ counts,
                            int* __restrict__ offsets,
                            int* __restrict__ cursors) {
  __shared__ int ssum[1024];
  int tid = threadIdx.x;
  int base = tid * 8;
  int lp[8];
  int run = 0;
#pragma unroll
  for (int j = 0; j < 8; ++j) { lp[j] = run; run += counts[base + j]; }
  ssum[tid] = run;
  __syncthreads();
  // Hillis-Steele inclusive scan over the 1024 partial sums
  for (int off = 1; off < 1024; off <<= 1) {
    int v = (tid >= off) ? ssum[tid - off] : 0;
    __syncthreads();
    ssum[tid] += v;
    __syncthreads();
  }
  int pref = (tid > 0) ? ssum[tid - 1] : 0;
#pragma unroll
  for (int j = 0; j < 8; ++j) {
    int o = pref + lp[j];
    offsets[base + j] = o;
    cursors[base + j] = o;
  }
}

// ---------------------------------------------------------------------------
// Kernel 4: scatter sentence ids into per-bag segments
// ---------------------------------------------------------------------------
__global__ void scatter_kernel(const long long* __restrict__ epid,
                               int* __restrict__ cursors,
                               int* __restrict__ sidx) {
  int s = blockIdx.x * blockDim.x + threadIdx.x;
  if (s < N_SENT) {
    int q = (int)epid[s];
    int pos = atomicAdd(&cursors[q], 1);
    sidx[pos] = s;
  }
}

// ---------------------------------------------------------------------------
// Kernel 5: WMMA fp32 GEMM  scores[N_SENT, 64] = inputs[N_SENT, D] x embPadT
// One wave per 16-ROW tile computes all 4 column tiles: the A fragment is
// loaded once per k-step and reused by 4 v_wmma_f32_16x16x4_f32 into 4
// independent accumulators (A traffic 181 MB instead of 724 MB; the 4-apart
// accumulator reuse also hides WMMA RAW latency). Main loop is unguarded
// (172 iters, K=0..687); a single guarded epilogue handles K=688..691
// (lanes 16-31 feed zeros for K=690,691; B side zero-padded to KPAD).
// Layouts per ISA 7.12.2:
//   A 16x4 f32: lanes 0-15 hold {K=k,k+1}, lanes 16-31 hold {K=k+2,k+3}
//   B 4x16 f32: same K split; N = lane & 15
//   C/D 16x16:  VGPR j -> M=j (lanes 0-15) / M=8+j (lanes 16-31)
// ---------------------------------------------------------------------------
__global__ void wmma_score_gemm(const float* __restrict__ inputs,
                                const float* __restrict__ embPad,
                                float* __restrict__ scores) {
  int lane = threadIdx.x & 31;
  int wave = (blockIdx.x * blockDim.x + threadIdx.x) >> 5;  // global wave id
  int tm   = wave;            // 0..4095 row tiles
  int half = lane >> 4;       // which K-pair this lane supplies
  int l15  = lane & 15;

  const float* arow = inputs + (size_t)(tm * 16 + l15) * D;
  const float* b0 = embPad + (size_t)(0 * 16 + l15) * KPAD;
  const float* b1 = embPad + (size_t)(1 * 16 + l15) * KPAD;
  const float* b2 = embPad + (size_t)(2 * 16 + l15) * KPAD;
  const float* b3 = embPad + (size_t)(3 * 16 + l15) * KPAD;

  v8f acc0 = {}, acc1 = {}, acc2 = {}, acc3 = {};
  int ko = 2 * half;

  // Main loop: 172 unguarded k-steps (K = 0..687).
  for (int k = 0; k < 688; k += 4) {
    int ka = k + ko;
    v2f a  = *(const v2f*)(arow + ka);
    v2f vb0 = *(const v2f*)(b0 + ka);
    v2f vb1 = *(const v2f*)(b1 + ka);
    v2f vb2 = *(const v2f*)(b2 + ka);
    v2f vb3 = *(const v2f*)(b3 + ka);
    acc0 = __builtin_amdgcn_wmma_f32_16x16x4_f32(false, a, false, vb0,
                                                 (short)0, acc0, false, false);
    acc1 = __builtin_amdgcn_wmma_f32_16x16x4_f32(false, a, false, vb1,
                                                 (short)0, acc1, false, false);
    acc2 = __builtin_amdgcn_wmma_f32_16x16x4_f32(false, a, false, vb2,
                                                 (short)0, acc2, false, false);
    acc3 = __builtin_amdgcn_wmma_f32_16x16x4_f32(false, a, false, vb3,
                                                 (short)0, acc3, false, false);
  }

  // Epilogue: K = 688..691 (690, 691 are out of range -> zeros on hi lanes).
  {
    const v2f zz = {0.0f, 0.0f};
    int ka = 688 + ko;
    v2f a  = (half == 0) ? *(const v2f*)(arow + ka) : zz;
    v2f vb0 = *(const v2f*)(b0 + ka);   // embPad rows zero-padded to KPAD
    v2f vb1 = *(const v2f*)(b1 + ka);
    v2f vb2 = *(const v2f*)(b2 + ka);
    v2f vb3 = *(const v2f*)(b3 + ka);
    acc0 = __builtin_amdgcn_wmma_f32_16x16x4_f32(false, a, false, vb0,
                                                 (short)0, acc0, false, false);
    acc1 = __builtin_amdgcn_wmma_f32_16x16x4_f32(false, a, false, vb1,
                                                 (short)0, acc1, false, false);
    acc2 = __builtin_amdgcn_wmma_f32_16x16x4_f32(false, a, false, vb2,
                                                 (short)0, acc2, false, false);
    acc3 = __builtin_amdgcn_wmma_f32_16x16x4_f32(false, a, false, vb3,
                                                 (short)0, acc3, false, false);
  }

  // Store: VGPR j -> row M = j + 8*half within the tile.
#pragma unroll
  for (int j = 0; j < 8; ++j) {
    size_t row = (size_t)(tm * 16 + j + 8 * half) * NPAD + l15;
    scores[row + 0]  = acc0[j];
    scores[row + 16] = acc1[j];
    scores[row + 32] = acc2[j];
    scores[row + 48] = acc3[j];
  }
}

// ---------------------------------------------------------------------------
// Kernel 6: per-bag selective attention. One 256-thread block per bag.
// out row p = N_PAIRS-1-q (reference reverses bincount order).
// alpha_i = exp(s_i - m) / (sum_j exp(s_j - m) + 1e-8)
// ---------------------------------------------------------------------------
__global__ void attn_kernel(const float* __restrict__ inputs,
                            const long long* __restrict__ labels,
                            const float* __restrict__ scores,
                            const int* __restrict__ counts,
                            const int* __restrict__ offsets,
                            const int* __restrict__ sidx,
                            float* __restrict__ out) {
  __shared__ int   s_sid[MAXC];
  __shared__ float s_w[MAXC];
  __shared__ int   s_lab;

  int q   = blockIdx.x;
  int tid = threadIdx.x;
  int cnt = counts[q];
  int off = offsets[q];
  int cc  = cnt < MAXC ? cnt : MAXC;
  int p   = N_PAIRS - 1 - q;

  for (int i = tid; i < cc; i += blockDim.x) s_sid[i] = sidx[off + i];
  __syncthreads();

  if (cc == 0) {  // cannot happen per reference setup; keep deterministic
    for (int d = tid; d < D; d += blockDim.x) out[(size_t)p * D + d] = 0.0f;
    if (tid == 0) out[(size_t)N_PAIRS * D + p] = 0.0f;
    return;
  }

  // Deterministic reduction order: sort bag members (tiny: avg 8, max ~25).
  if (tid == 0) {
    for (int i = 1; i < cc; ++i) {
      int v = s_sid[i], j = i - 1;
      while (j >= 0 && s_sid[j] > v) { s_sid[j + 1] = s_sid[j]; --j; }
      s_sid[j + 1] = v;
    }
    s_lab = (int)labels[s_sid[0]];  // labels consistent within a bag
  }
  __syncthreads();
  int lab = s_lab;

  // Gather this bag's scores (from the WMMA GEMM result).
  for (int i = tid; i < cc; i += blockDim.x)
    s_w[i] = scores[(size_t)s_sid[i] * NPAD + lab];
  __syncthreads();

  // Every thread computes m and S serially from LDS (cc is tiny).
  float m = -3.4e38f;
  for (int i = 0; i < cc; ++i) m = fmaxf(m, s_w[i]);
  float S = 0.0f;
  for (int i = 0; i < cc; ++i) S += __expf(s_w[i] - m);
  float inv = 1.0f / (S + 1e-8f);
  __syncthreads();
  for (int i = tid; i < cc; i += blockDim.x) s_w[i] = __expf(s_w[i] - m);
  __syncthreads();

  // Warm the bag rows (global_prefetch_b8), then weighted sum over columns.
  for (int i = tid; i < cc; i += blockDim.x)
    __builtin_prefetch(inputs + (size_t)s_sid[i] * D, 0, 0);

  for (int d = tid; d < D; d += blockDim.x) {
    float acc = 0.0f;
    for (int i = 0; i < cc; ++i)
      acc += s_w[i] * inputs[(size_t)s_sid[i] * D + d];
    out[(size_t)p * D + d] = acc * inv;
  }
  if (tid == 0) out[(size_t)N_PAIRS * D + p] = (float)lab;
}

// ---------------------------------------------------------------------------
extern "C" void kernel_launch(void* const* d_in, const int* in_sizes, int n_in,
                              void* d_out, int out_size, void* d_ws, size_t ws_size,
                              hipStream_t stream) {
  const float*     inputs = (const float*)d_in[0];      // [N_SENT, D] f32
  const float*     emb    = (const float*)d_in[1];      // [CLS, D]    f32
  const long long* labels = (const long long*)d_in[2];  // [N_SENT]    i64
  const long long* epid   = (const long long*)d_in[3];  // [N_SENT]    i64
  float* out = (float*)d_out;

  // Workspace partition (~17.3 MB total).
  float* embPad  = (float*)d_ws;                         // 64*704 floats
  float* scores  = embPad + NPAD * KPAD;                 // 65536*64 floats
  int*   counts  = (int*)(scores + (size_t)N_SENT * NPAD);
  int*   offsets = counts + N_PAIRS;
  int*   cursors = offsets + N_PAIRS;
  int*   sidx    = cursors + N_PAIRS;                    // N_SENT ints

  const int T = 256;
  prep_kernel<<<(NPAD * KPAD + T - 1) / T, T, 0, stream>>>(emb, embPad, counts);
  count_kernel<<<N_SENT / T, T, 0, stream>>>(epid, counts);
  scan_kernel<<<1, 1024, 0, stream>>>(counts, offsets, cursors);
  scatter_kernel<<<N_SENT / T, T, 0, stream>>>(epid, cursors, sidx);
  // 4096 row-tile waves; 8 waves per block -> 512 blocks.
  wmma_score_gemm<<<512, T, 0, stream>>>(inputs, embPad, scores);
  attn_kernel<<<N_PAIRS, T, 0, stream>>>(inputs, labels, scores,
                                         counts, offsets, sidx, out);
}